// Net_39298950758475
// MI455X (gfx1250) — compile-verified
//
#include <hip/hip_runtime.h>
#include <math.h>

// ---------------------------------------------------------------------------
// FFF net: tree routing + per-leaf MLP + softmax.
// Memory-bound (~150MB traffic @ 23.3TB/s ~ 6.5us floor). Strategy: bin
// samples by leaf so each leaf's W1/W2 (512KB) streams exactly once, and run
// the small dense GEMMs on the matrix pipe with fp32 WMMA (16x16x4).
// ---------------------------------------------------------------------------

#define INPUT_W   1024
#define LEAF_H    64
#define OUT_W     1000
#define DEPTH     8
#define N_LEAVES  256
#define N_NODES   255
#define BATCH     2048
#define LEAF_CAP  2048          // worst case: every sample lands in one leaf
#define NTILES_O  63            // ceil(1000/16)

typedef __attribute__((ext_vector_type(2))) float v2f;
typedef __attribute__((ext_vector_type(8))) float v8f;

// ---------------------------------------------------------------------------
// Kernel 0: zero per-leaf counters (must run every call: deterministic)
// ---------------------------------------------------------------------------
__global__ __launch_bounds__(256) void fff_zero_counts(int* __restrict__ count) {
    count[threadIdx.x] = 0;
}

// ---------------------------------------------------------------------------
// Kernel 1: hard tree descent. One wave (32 lanes) per sample. The x-row is
// cached in 32 VGPRs per lane and reused across all 8 levels, so x is read
// once from HBM. Node weights (1MB total) live in L2 after first touch.
// ---------------------------------------------------------------------------
__global__ __launch_bounds__(256) void fff_route(
    const float* __restrict__ x,
    const float* __restrict__ nodeW,
    const float* __restrict__ nodeB,
    int* __restrict__ count,
    int* __restrict__ bins)
{
    const int wave = threadIdx.x >> 5;
    const int lane = threadIdx.x & 31;
    const int sample = blockIdx.x * 8 + wave;          // grid = BATCH/8

    const float4* xr = (const float4*)(x + sample * INPUT_W);
    float xs[32];
    #pragma unroll
    for (int i = 0; i < 8; ++i) {
        float4 v = xr[lane + i * 32];                  // 8 float4 per lane
        xs[4*i+0] = v.x; xs[4*i+1] = v.y; xs[4*i+2] = v.z; xs[4*i+3] = v.w;
    }

    int cur = 0;
    #pragma unroll
    for (int d = 0; d < DEPTH; ++d) {
        const float4* w = (const float4*)(nodeW + cur * INPUT_W);
        float acc = 0.f;
        #pragma unroll
        for (int i = 0; i < 8; ++i) {
            float4 wv = w[lane + i * 32];
            acc += xs[4*i+0]*wv.x + xs[4*i+1]*wv.y + xs[4*i+2]*wv.z + xs[4*i+3]*wv.w;
        }
        // full wave32 butterfly reduce -> every lane has the total
        #pragma unroll
        for (int off = 16; off >= 1; off >>= 1)
            acc += __shfl_xor(acc, off, 32);
        float score = acc + nodeB[cur];
        cur = 2 * cur + 1 + (score >= 0.f ? 1 : 0);    // uniform across wave
    }

    if (lane == 0) {
        int leaf = cur - N_NODES;
        int pos = atomicAdd(&count[leaf], 1);
        bins[leaf * LEAF_CAP + pos] = sample;
    }
}

// ---------------------------------------------------------------------------
// Kernel 2: fused per-leaf MLP on the matrix pipe.
// Block = 128 threads (4 waves), handles one 16-sample m-tile of one leaf.
//   GEMM1: h(16x64) = x(16x1024) @ W1[leaf](1024x64), wave w owns cols 16w..
//   relu+bias -> LDS, then GEMM2 over 63 output tiles of 16, scatter logits.
// fp32 WMMA keeps reference precision; compute is far below the BW roofline
// so there is nothing to gain from low precision here.
// ---------------------------------------------------------------------------
__global__ __launch_bounds__(128) void fff_leaf_mlp(
    const float* __restrict__ x,
    const float* __restrict__ w1s,
    const float* __restrict__ b1s,
    const float* __restrict__ w2s,
    const float* __restrict__ b2s,
    const int* __restrict__ count,
    const int* __restrict__ bins,
    float* __restrict__ out)
{
    const int leaf = blockIdx.x;                        // 0..255
    const int tile = blockIdx.y;                        // 0..127
    const int rows = count[leaf];
    const int r0 = tile * 16;
    if (r0 >= rows) return;                             // uniform early-exit
    const int nvalid = min(16, rows - r0);

    const int wave  = threadIdx.x >> 5;                 // 0..3
    const int lane  = threadIdx.x & 31;
    const int m     = lane & 15;                        // row within tile / col within tile
    const int khalf = (lane >> 4) * 2;                  // 0 (lanes 0-15) or 2 (lanes 16-31)

    __shared__ float h_lds[16 * LEAF_H];                // 4KB hidden tile
    __shared__ int   rowidx[16];

    if (threadIdx.x < 16) {
        int rr = min((int)threadIdx.x, nvalid - 1);     // clamp-pad short tiles
        rowidx[threadIdx.x] = bins[leaf * LEAF_CAP + r0 + rr];
    }
    __syncthreads();

    // ---- GEMM1: K = 1024 in steps of 4 ----
    const float* xrow = x + (size_t)rowidx[m] * INPUT_W;   // A: M = lane&15
    const float* w1   = w1s + (size_t)leaf * INPUT_W * LEAF_H;
    const int h0 = wave * 16;                              // this wave's N tile

    v8f c1 = {};
    #pragma unroll 4
    for (int k = 0; k < INPUT_W; k += 4) {
        v2f a, b;
        a.x = xrow[k + khalf];                             // K = k+khalf
        a.y = xrow[k + khalf + 1];                         // K = k+khalf+1
        b.x = w1[(size_t)(k + khalf    ) * LEAF_H + h0 + m];
        b.y = w1[(size_t)(k + khalf + 1) * LEAF_H + h0 + m];
        c1 = __builtin_amdgcn_wmma_f32_16x16x4_f32(false, a, false, b,
                                                   (short)0, c1, false, false);
    }

    // bias + relu, write hidden tile to LDS (C layout: M = r + 8*(lane>=16))
    const float bias1 = b1s[leaf * LEAF_H + h0 + m];
    #pragma unroll
    for (int r = 0; r < 8; ++r) {
        int M = r + (lane >> 4) * 8;
        float v = c1[r] + bias1;
        h_lds[M * LEAF_H + h0 + m] = v > 0.f ? v : 0.f;
    }
    __syncthreads();

    // ---- GEMM2: 63 output tiles of 16 over 4 waves, K = 64 in steps of 4 ----
    const float* w2 = w2s + (size_t)leaf * LEAF_H * OUT_W;
    for (int nt = wave; nt < NTILES_O; nt += 4) {
        const int o0 = nt * 16;
        const int o  = o0 + m;
        const bool ocol = (o < OUT_W);
        v8f c2 = {};
        #pragma unroll
        for (int k = 0; k < LEAF_H; k += 4) {
            v2f a, b;
            a.x = h_lds[m * LEAF_H + k + khalf];
            a.y = h_lds[m * LEAF_H + k + khalf + 1];
            b.x = ocol ? w2[(size_t)(k + khalf    ) * OUT_W + o] : 0.f;
            b.y = ocol ? w2[(size_t)(k + khalf + 1) * OUT_W + o] : 0.f;
            c2 = __builtin_amdgcn_wmma_f32_16x16x4_f32(false, a, false, b,
                                                       (short)0, c2, false, false);
        }
        const float bias2 = ocol ? b2s[leaf * OUT_W + o] : 0.f;
        #pragma unroll
        for (int r = 0; r < 8; ++r) {
            int M = r + (lane >> 4) * 8;
            if (M < nvalid && ocol)
                out[(size_t)rowidx[M] * OUT_W + o] = c2[r] + bias2;
        }
    }
}

// ---------------------------------------------------------------------------
// Kernel 3: in-place row softmax over d_out (2048 rows x 1000)
// ---------------------------------------------------------------------------
__global__ __launch_bounds__(256) void fff_softmax(float* __restrict__ out) {
    const int row = blockIdx.x;
    float* p = out + (size_t)row * OUT_W;
    const int tid = threadIdx.x;
    __shared__ float red[8];

    float mx = -INFINITY;
    for (int i = tid; i < OUT_W; i += 256) mx = fmaxf(mx, p[i]);
    #pragma unroll
    for (int off = 16; off >= 1; off >>= 1) mx = fmaxf(mx, __shfl_xor(mx, off, 32));
    if ((tid & 31) == 0) red[tid >> 5] = mx;
    __syncthreads();
    if (tid == 0) {
        float v = red[0];
        #pragma unroll
        for (int i = 1; i < 8; ++i) v = fmaxf(v, red[i]);
        red[0] = v;
    }
    __syncthreads();
    mx = red[0];

    float s = 0.f;
    for (int i = tid; i < OUT_W; i += 256) {
        float e = __expf(p[i] - mx);
        p[i] = e;
        s += e;
    }
    #pragma unroll
    for (int off = 16; off >= 1; off >>= 1) s += __shfl_xor(s, off, 32);
    __shared__ float red2[8];
    if ((tid & 31) == 0) red2[tid >> 5] = s;
    __syncthreads();
    if (tid == 0) {
        float t = 0.f;
        #pragma unroll
        for (int i = 0; i < 8; ++i) t += red2[i];
        red2[0] = t;
    }
    __syncthreads();
    const float inv = 1.0f / red2[0];
    for (int i = tid; i < OUT_W; i += 256) p[i] *= inv;
}

// ---------------------------------------------------------------------------
extern "C" void kernel_launch(void* const* d_in, const int* in_sizes, int n_in,
                              void* d_out, int out_size, void* d_ws, size_t ws_size,
                              hipStream_t stream) {
    const float* x     = (const float*)d_in[0];
    const float* nodeW = (const float*)d_in[1];
    const float* nodeB = (const float*)d_in[2];
    const float* w1s   = (const float*)d_in[3];
    const float* b1s   = (const float*)d_in[4];
    const float* w2s   = (const float*)d_in[5];
    const float* b2s   = (const float*)d_in[6];
    float* out = (float*)d_out;

    // ws layout: [0,256) leaf counters; then 256*2048 binned sample ids (~2.1MB)
    int* count = (int*)d_ws;
    int* bins  = count + N_LEAVES;

    fff_zero_counts<<<1, 256, 0, stream>>>(count);
    fff_route<<<BATCH / 8, 256, 0, stream>>>(x, nodeW, nodeB, count, bins);
    fff_leaf_mlp<<<dim3(N_LEAVES, BATCH / 16), 128, 0, stream>>>(
        x, w1s, b1s, w2s, b2s, count, bins, out);
    fff_softmax<<<BATCH, 256, 0, stream>>>(out);
}